// GraphSAGEModel_8117488189609
// MI455X (gfx1250) — compile-verified
//
#include <hip/hip_runtime.h>
#include <hip/hip_bf16.h>

typedef __attribute__((ext_vector_type(2))) float v2f;
typedef __attribute__((ext_vector_type(8))) float v8f;

#define N_NODES   8192
#define TILE_K    256
#define LDS_STRIDE 260   // 1040B rows: 16B aligned, conflict-free (4*m bank skew)
#define BN_EPS    1e-5f

// ---------------------------------------------------------------------------
// zT[n2][k] = ( optional BN(in[k]) ) @ W   (projection, transposed output)
// din <= 18, dout <= 16; rows n2 >= dout are zero-filled (padding).
// stats = {mu[16], var[16]} from previous layer; use_bn=0 -> raw input.
// ---------------------------------------------------------------------------
__global__ __launch_bounds__(256)
void zproj_kernel(const float* __restrict__ in, const float* __restrict__ W,
                  const float* __restrict__ gamma, const float* __restrict__ beta,
                  const float* __restrict__ stats,
                  float* __restrict__ zT, int din, int dout, int use_bn)
{
    int k = blockIdx.x * blockDim.x + threadIdx.x;
    if (k >= N_NODES) return;
    float v[20];
    const float* row = in + (size_t)k * din;
    for (int n = 0; n < din; ++n) {
        float xv = row[n];
        if (use_bn) {
            float a = gamma[n] * rsqrtf(stats[16 + n] + BN_EPS);
            xv = a * (xv - stats[n]) + beta[n];
        }
        v[n] = xv;
    }
    for (int n2 = 0; n2 < 16; ++n2) {
        float acc = 0.f;
        if (n2 < dout) {
            for (int n = 0; n < din; ++n)
                acc += v[n] * W[n * dout + n2];
        }
        zT[(size_t)n2 * N_NODES + k] = acc;
    }
}

// ---------------------------------------------------------------------------
// h[8192][16] = relu(adj @ z + b), z given transposed as zT[16][8192].
// One workgroup (8 waves) per 16-row tile; fp32 WMMA 16x16x4; waves split K,
// partials reduced through LDS. Fully deterministic (no atomics).
// ---------------------------------------------------------------------------
__global__ __launch_bounds__(256)
void sage_gemm_kernel(const float* __restrict__ adj, const float* __restrict__ zT,
                      const float* __restrict__ bias, float* __restrict__ h, int dout)
{
    __shared__ __align__(16) float lds_a[16 * LDS_STRIDE];
    __shared__ __align__(16) float lds_b[16 * LDS_STRIDE];
    __shared__ float red[8 * 256];

    const int t    = threadIdx.x;          // 0..255
    const int tile = blockIdx.x;           // 0..511
    const int wv   = t >> 5;               // wave id 0..7
    const int lane = t & 31;
    const int m    = lane & 15;            // row within tile / B column index
    const int koff = (lane >> 4) << 1;     // 0 or 2 (half-wave K offset)

    v8f c = {};                            // 16x16 fp32 accumulator (8 VGPRs)

    const size_t rowbase = (size_t)tile * 16 * N_NODES;

    for (int k0 = 0; k0 < N_NODES; k0 += TILE_K) {
        // Cooperative coalesced load: adj tile 16x256 and zT tile 16x256.
        #pragma unroll
        for (int i = 0; i < 4; ++i) {
            int idx = i * 256 + t;              // 0..1023 float4 slots
            int r   = idx >> 6;                 // row 0..15
            int c4  = (idx & 63) << 2;          // col 0..252 step 4
            float4 va = *(const float4*)(adj + rowbase + (size_t)r * N_NODES + k0 + c4);
            *(float4*)&lds_a[r * LDS_STRIDE + c4] = va;
            float4 vb = *(const float4*)(zT + (size_t)r * N_NODES + k0 + c4);
            *(float4*)&lds_b[r * LDS_STRIDE + c4] = vb;
        }
        __syncthreads();

        // Wave wv covers K sub-range [wv*32, wv*32+32) of this chunk: 8 WMMAs.
        const float* pa = &lds_a[m * LDS_STRIDE + wv * 32 + koff];
        const float* pb = &lds_b[m * LDS_STRIDE + wv * 32 + koff];
        #pragma unroll
        for (int s = 0; s < 8; ++s) {
            v2f a = *(const v2f*)(pa + s * 4);   // {K=k, K=k+1} per ISA A layout
            v2f b = *(const v2f*)(pb + s * 4);   // {K=k, K=k+1} per ISA B layout
            c = __builtin_amdgcn_wmma_f32_16x16x4_f32(
                    false, a, false, b, (short)0, c, false, false);
        }
        __syncthreads();
    }

    // Cross-wave reduction of partial C tiles.
    #pragma unroll
    for (int v = 0; v < 8; ++v) red[wv * 256 + lane * 8 + v] = c[v];
    __syncthreads();
    float s = 0.f;
    #pragma unroll
    for (int w = 0; w < 8; ++w) s += red[w * 256 + t];

    // Slot t -> (lane' = t>>3, vgpr = t&7); C layout: M = v + 8*(lane'>>4), N = lane'&15.
    int vv = t & 7, slot = t >> 3;
    int mm = vv + ((slot >> 4) << 3);
    int nn = slot & 15;
    float bv = (nn < dout) ? bias[nn] : 0.f;
    float hv = fmaxf(s + bv, 0.f);
    h[((size_t)tile * 16 + mm) * 16 + nn] = hv;
}

// ---------------------------------------------------------------------------
// Per-column biased batch stats: stats = {mu[16], var[16]}.
// ---------------------------------------------------------------------------
__global__ __launch_bounds__(256)
void stats_kernel(const float* __restrict__ h, float* __restrict__ stats)
{
    __shared__ float ss[256], ss2[256];
    int n = blockIdx.x, t = threadIdx.x;
    float s = 0.f, s2 = 0.f;
    for (int r = t; r < N_NODES; r += 256) {
        float v = h[(size_t)r * 16 + n];
        s += v; s2 += v * v;
    }
    ss[t] = s; ss2[t] = s2;
    __syncthreads();
    for (int o = 128; o > 0; o >>= 1) {
        if (t < o) { ss[t] += ss[t + o]; ss2[t] += ss2[t + o]; }
        __syncthreads();
    }
    if (t == 0) {
        float mu = ss[0] * (1.0f / N_NODES);
        stats[n]      = mu;
        stats[16 + n] = ss2[0] * (1.0f / N_NODES) - mu * mu;
    }
}

// ---------------------------------------------------------------------------
// out[n] = mean_k( g*(h[k][n]-mu)*rsqrt(var+eps) + be ),  n in 0..7
// ---------------------------------------------------------------------------
__global__ __launch_bounds__(256)
void final_kernel(const float* __restrict__ h, const float* __restrict__ stats,
                  const float* __restrict__ gamma, const float* __restrict__ beta,
                  float* __restrict__ out)
{
    __shared__ float ss[256];
    int n = blockIdx.x, t = threadIdx.x;
    float mu = stats[n];
    float a  = gamma[n] * rsqrtf(stats[16 + n] + BN_EPS);
    float be = beta[n];
    float s = 0.f;
    for (int r = t; r < N_NODES; r += 256)
        s += a * (h[(size_t)r * 16 + n] - mu) + be;
    ss[t] = s;
    __syncthreads();
    for (int o = 128; o > 0; o >>= 1) {
        if (t < o) ss[t] += ss[t + o];
        __syncthreads();
    }
    if (t == 0) out[n] = ss[0] * (1.0f / N_NODES);
}

// ---------------------------------------------------------------------------
extern "C" void kernel_launch(void* const* d_in, const int* in_sizes, int n_in,
                              void* d_out, int out_size, void* d_ws, size_t ws_size,
                              hipStream_t stream)
{
    const float* x   = (const float*)d_in[0];
    const float* adj = (const float*)d_in[1];
    const float* W1  = (const float*)d_in[2];
    const float* b1  = (const float*)d_in[3];
    const float* g1  = (const float*)d_in[4];
    const float* be1 = (const float*)d_in[5];
    const float* W2  = (const float*)d_in[6];
    const float* b2  = (const float*)d_in[7];
    const float* g2  = (const float*)d_in[8];
    const float* be2 = (const float*)d_in[9];
    const float* W3  = (const float*)d_in[10];
    const float* b3  = (const float*)d_in[11];
    const float* g3  = (const float*)d_in[12];
    const float* be3 = (const float*)d_in[13];
    float* out = (float*)d_out;

    char* ws = (char*)d_ws;
    const size_t SZ = (size_t)N_NODES * 16 * sizeof(float); // 512 KB
    float* zT    = (float*)(ws);            // [16][8192]
    float* hA    = (float*)(ws + SZ);       // [8192][16]
    float* hB    = (float*)(ws + 2 * SZ);   // [8192][16]
    float* stats = (float*)(ws + 3 * SZ);   // mu[16], var[16]

    dim3 blk(256);

    // Layer 1: z = x @ W1 ; h1 = relu(adj@z + b1)
    zproj_kernel    <<<N_NODES / 256, blk, 0, stream>>>(x,  W1, nullptr, nullptr, nullptr, zT, 18, 16, 0);
    sage_gemm_kernel<<<N_NODES / 16,  blk, 0, stream>>>(adj, zT, b1, hA, 16);
    stats_kernel    <<<16,            blk, 0, stream>>>(hA, stats);

    // Layer 2: z = bn1(h1) @ W2 ; h2 = relu(adj@z + b2)
    zproj_kernel    <<<N_NODES / 256, blk, 0, stream>>>(hA, W2, g1, be1, stats, zT, 16, 16, 1);
    sage_gemm_kernel<<<N_NODES / 16,  blk, 0, stream>>>(adj, zT, b2, hB, 16);
    stats_kernel    <<<16,            blk, 0, stream>>>(hB, stats);

    // Layer 3: z = bn2(h2) @ W3 (dout=8, padded) ; h3 = relu(adj@z + b3)
    zproj_kernel    <<<N_NODES / 256, blk, 0, stream>>>(hB, W3, g2, be2, stats, zT, 16, 8, 1);
    sage_gemm_kernel<<<N_NODES / 16,  blk, 0, stream>>>(adj, zT, b3, hA, 8);
    stats_kernel    <<<16,            blk, 0, stream>>>(hA, stats);

    // Final: mean over nodes of bn3(h3), columns 0..7
    final_kernel    <<<8,             blk, 0, stream>>>(hA, stats, g3, be3, out);
}